// GraphSAGE_62277025792168
// MI455X (gfx1250) — compile-verified
//
#include <hip/hip_runtime.h>
#include <hip/hip_bf16.h>
#include <math.h>

#define HID 64

typedef float v2f __attribute__((ext_vector_type(2)));
typedef float v8f __attribute__((ext_vector_type(8)));

// ---------------------------------------------------------------- zero (float4)
__global__ __launch_bounds__(256) void zero_f4(float4* __restrict__ p, long n4) {
    long i = (long)blockIdx.x * blockDim.x + threadIdx.x;
    if (i < n4) p[i] = make_float4(0.f, 0.f, 0.f, 0.f);
}

// ---------------------------------------------------------------- degree count
__global__ __launch_bounds__(256) void deg_count(const int* __restrict__ dst,
                                                 float* __restrict__ deg, int n_edges) {
    int e = blockIdx.x * blockDim.x + threadIdx.x;
    if (e < n_edges) atomicAdd(&deg[dst[e]], 1.0f);
}

__global__ __launch_bounds__(256) void deg_inv(float* __restrict__ deg, int n) {
    int i = blockIdx.x * blockDim.x + threadIdx.x;
    if (i < n) deg[i] = 1.0f / fmaxf(deg[i], 1.0f);
}

// ---------------------------------------------------------------- edge scatter
// one thread per (edge, 4-feature chunk): gather float4 from h[src], 4 atomic adds into msg[dst]
__global__ __launch_bounds__(256) void sage_scatter(const int* __restrict__ src,
                                                    const int* __restrict__ dst,
                                                    const float* __restrict__ h,
                                                    float* __restrict__ msg, int n_edges) {
    long gid = (long)blockIdx.x * blockDim.x + threadIdx.x;
    long e = gid >> 4;
    if (e >= n_edges) return;
    int f = (int)(gid & 15) * 4;
    int s = src[e], d = dst[e];
    float4 v = *(const float4*)(h + (size_t)s * HID + f);
    float* p = msg + (size_t)d * HID + f;
    atomicAdd(p + 0, v.x);
    atomicAdd(p + 1, v.y);
    atomicAdd(p + 2, v.z);
    atomicAdd(p + 3, v.w);
}

// ---------------------------------------------------------------- WMMA layer GEMM
// out[16x64 tile] = relu( (msg*inv_deg) @ Wl + x @ Wr + bl ), fused into one accumulator set.
// FP32 WMMA 16x16x4: A frag per lane = {A[m][kb], A[m][kb+1]}, m=lane&15, kb=k0+2*(lane>>4).
// B frag per lane = {W[kb][n], W[kb+1][n]}, n=n0+(lane&15). Weights staged in LDS.
__global__ __launch_bounds__(256) void sage_gemm(const float* __restrict__ msg,
                                                 const float* __restrict__ inv_deg,
                                                 const float* __restrict__ h_in,
                                                 const float* __restrict__ Wl,
                                                 const float* __restrict__ bl,
                                                 const float* __restrict__ Wr,
                                                 float* __restrict__ h_out, int n_nodes) {
    __shared__ float sW[2 * HID * HID];  // [0..4095]=Wl, [4096..8191]=Wr (row-major k,n)
    {
        const float4* wl4 = (const float4*)Wl;
        const float4* wr4 = (const float4*)Wr;
        float4* s4 = (float4*)sW;
        const int Q = HID * HID / 4;  // 1024 float4 per matrix
        for (int i = threadIdx.x; i < Q; i += 256) {
            s4[i] = wl4[i];
            s4[i + Q] = wr4[i];
        }
    }
    __syncthreads();

    int wave = (int)((blockIdx.x * 256 + threadIdx.x) >> 5);  // m-tile index
    int lane = threadIdx.x & 31;
    int row0 = wave * 16;
    if (row0 >= n_nodes) return;  // uniform per-wave exit: EXEC stays all-ones for WMMA

    const int m = lane & 15;
    const int kh = (lane >> 4) * 2;   // K sub-offset for this lane half
    const int ncol = lane & 15;       // B/N column within tile

    const float* mrow = msg + (size_t)(row0 + m) * HID;
    const float* xrow = h_in + (size_t)(row0 + m) * HID;
    const float inv = inv_deg[row0 + m];

    v8f acc[4] = {};  // 4 n-tiles of 16 -> full 64 output columns

    for (int k0 = 0; k0 < HID; k0 += 4) {
        int kb = k0 + kh;
        v2f am, ax;
        am.x = mrow[kb] * inv;
        am.y = mrow[kb + 1] * inv;
        ax.x = xrow[kb];
        ax.y = xrow[kb + 1];
#pragma unroll
        for (int nt = 0; nt < 4; ++nt) {
            int n = nt * 16 + ncol;
            v2f bL, bR;
            bL.x = sW[kb * HID + n];
            bL.y = sW[(kb + 1) * HID + n];
            bR.x = sW[HID * HID + kb * HID + n];
            bR.y = sW[HID * HID + (kb + 1) * HID + n];
            acc[nt] = __builtin_amdgcn_wmma_f32_16x16x4_f32(
                false, am, false, bL, (short)0, acc[nt], false, false);
            acc[nt] = __builtin_amdgcn_wmma_f32_16x16x4_f32(
                false, ax, false, bR, (short)0, acc[nt], false, false);
        }
    }

    // D layout: reg r holds (m = r + 8*(lane>>4), n = nt*16 + (lane&15))
    const int mbase = (lane >> 4) * 8;
#pragma unroll
    for (int nt = 0; nt < 4; ++nt) {
        int nn = nt * 16 + (lane & 15);
        float b = bl[nn];
#pragma unroll
        for (int r = 0; r < 8; ++r) {
            int mm = mbase + r;
            float v = acc[nt][r] + b;
            v = v > 0.0f ? v : 0.0f;
            h_out[(size_t)(row0 + mm) * HID + nn] = v;
        }
    }
}

// ---------------------------------------------------------------- final FC + sigmoid
// one wave per graph: dot(h[g, 0:320], Wfc) via strided lanes + shfl_xor reduce
__global__ __launch_bounds__(256) void fc_sigmoid(const float* __restrict__ h,
                                                  const float* __restrict__ Wfc,
                                                  const float* __restrict__ bfc,
                                                  float* __restrict__ out, int batch) {
    int wave = (int)((blockIdx.x * 256 + threadIdx.x) >> 5);
    int lane = threadIdx.x & 31;
    if (wave >= batch) return;
    const float* row = h + (size_t)wave * (5 * HID);
    float s = 0.f;
#pragma unroll
    for (int k = lane; k < 5 * HID; k += 32) s += row[k] * Wfc[k];
#pragma unroll
    for (int off = 16; off; off >>= 1) s += __shfl_xor(s, off, 32);
    if (lane == 0) {
        float z = s + bfc[0];
        out[wave] = 1.0f / (1.0f + __expf(-z));
    }
}

// ---------------------------------------------------------------- launch
extern "C" void kernel_launch(void* const* d_in, const int* in_sizes, int n_in,
                              void* d_out, int out_size, void* d_ws, size_t ws_size,
                              hipStream_t stream) {
    const float* x   = (const float*)d_in[0];
    const int*   ei  = (const int*)d_in[1];
    const float* Wl  = (const float*)d_in[2];
    const float* bl  = (const float*)d_in[3];
    const float* Wr  = (const float*)d_in[4];
    const float* Wfc = (const float*)d_in[5];
    const float* bfc = (const float*)d_in[6];

    const int n_nodes = in_sizes[0] / HID;   // 100000
    const int n_edges = in_sizes[1] / 2;     // 1600000
    const int* src = ei;
    const int* dst = ei + n_edges;

    // workspace layout (bytes): inv_deg | msg | hA | hB   (all 16B-aligned)
    float* invdeg = (float*)d_ws;
    float* msg = invdeg + n_nodes;               // n_nodes*4 = 400000 B, %16==0
    float* hA = msg + (size_t)n_nodes * HID;
    float* hB = hA + (size_t)n_nodes * HID;

    // --- degrees (layer-invariant): deg -> 1/max(deg,1), stored in invdeg
    zero_f4<<<(n_nodes / 4 + 255) / 256, 256, 0, stream>>>((float4*)invdeg, n_nodes / 4);
    deg_count<<<(n_edges + 255) / 256, 256, 0, stream>>>(dst, invdeg, n_edges);
    deg_inv<<<(n_nodes + 255) / 256, 256, 0, stream>>>(invdeg, n_nodes);

    const long msg_f4 = (long)n_nodes * HID / 4;
    const long scat_threads = (long)n_edges * 16;
    const int gemm_blocks = (n_nodes / 16 + 7) / 8;  // 8 waves (m-tiles) per block

    const float* hin = x;
    float* houts[5] = {hA, hB, hA, hB, hA};
    for (int l = 0; l < 5; ++l) {
        zero_f4<<<(int)((msg_f4 + 255) / 256), 256, 0, stream>>>((float4*)msg, msg_f4);
        sage_scatter<<<(int)((scat_threads + 255) / 256), 256, 0, stream>>>(
            src, dst, hin, msg, n_edges);
        sage_gemm<<<gemm_blocks, 256, 0, stream>>>(
            msg, invdeg, hin, Wl + (size_t)l * HID * HID, bl + (size_t)l * HID,
            Wr + (size_t)l * HID * HID, houts[l], n_nodes);
        hin = houts[l];
    }

    const int batch = n_nodes / 5;  // 20000
    fc_sigmoid<<<(batch * 32 + 255) / 256, 256, 0, stream>>>(hA, Wfc, bfc,
                                                             (float*)d_out, batch);
}